// SimpleGRU_35072702939659
// MI455X (gfx1250) — compile-verified
//
#include <hip/hip_runtime.h>

typedef __attribute__((ext_vector_type(2))) float v2f;
typedef __attribute__((ext_vector_type(8))) float v8f;

#define GRU_T 512
#define GRU_H 16

__device__ __forceinline__ float sigmoid_fast(float x) {
    float e = __expf(-x);
    return __builtin_amdgcn_rcpf(1.0f + e);
}
__device__ __forceinline__ float tanh_fast(float x) {
    float e = __expf(-2.0f * x);
    return (1.0f - e) * __builtin_amdgcn_rcpf(1.0f + e);
}

// One wave32 per 16-batch tile. Recurrent 16x16x16 f32 GEMM per gate done as
// 4 chained V_WMMA_F32_16X16X4_F32. State h lives in the WMMA C/D layout
// (lane = batch column, VGPR v + 8*half = hidden index).
__global__ __launch_bounds__(32)
void gru_wmma_kernel(const float* __restrict__ x,
                     const float* __restrict__ W_ih,
                     const float* __restrict__ W_hh,
                     const float* __restrict__ b_ih,
                     const float* __restrict__ b_hh,
                     const float* __restrict__ fc_w,
                     const float* __restrict__ fc_b,
                     float* __restrict__ out)
{
    __shared__ float hbuf[16 * 18];   // h[batch][k], stride 18 dwords (8B-aligned pairs)
    __shared__ float red[32];

    const int lane = threadIdx.x;     // 0..31
    const int col  = lane & 15;       // batch-in-tile (WMMA N) == row M for A operand
    const int half = lane >> 4;       // lane half selects K / M sub-range
    const int b0   = blockIdx.x * 16;

    // ---- WMMA A operands: W_hh for 3 gates x 4 K-slices (constant) ----
    // A slice j, reg p, lane l: W_hh[(g*16 + l%16)*16 + 4j + p + 2*(l/16)]
    v2f A[3][4];
    #pragma unroll
    for (int g = 0; g < 3; ++g)
        #pragma unroll
        for (int j = 0; j < 4; ++j)
            A[g][j] = *(const v2f*)(W_hh + (g * 16 + col) * 16 + 4 * j + 2 * half);

    // ---- per-lane gate constants (n = v + 8*half in C/D layout) ----
    float wr[8], wz[8], wn[8], cr0[8], cz0[8], cn0[8], bnc[8], fw[8];
    #pragma unroll
    for (int v = 0; v < 8; ++v) {
        int n = v + 8 * half;
        wr[v]  = W_ih[n];
        wz[v]  = W_ih[16 + n];
        wn[v]  = W_ih[32 + n];
        cr0[v] = b_ih[n]      + b_hh[n];        // fold both biases for r
        cz0[v] = b_ih[16 + n] + b_hh[16 + n];   // fold both biases for z
        cn0[v] = b_hh[32 + n];                  // hh-bias only for n gate
        bnc[v] = b_ih[32 + n];                  // ih-bias for n gate (outside r*)
        fw[v]  = fc_w[n];
    }

    float h[8];
    #pragma unroll
    for (int v = 0; v < 8; ++v) h[v] = 0.0f;

    const float* xrow = x + (size_t)(b0 + col) * GRU_T;

    // prefetch 4 timesteps of x (I == 1, so x[b,t] is a scalar stream per batch)
    float4 xq = *(const float4*)(xrow);

    for (int t4 = 0; t4 < GRU_T / 4; ++t4) {
        float4 xcur = xq;
        if (t4 + 1 < GRU_T / 4) {
            xq = *(const float4*)(xrow + 4 * (t4 + 1));
            __builtin_prefetch(xrow + 4 * (t4 + 2), 0, 0);
        }
        float xs[4] = {xcur.x, xcur.y, xcur.z, xcur.w};

        #pragma unroll
        for (int s = 0; s < 4; ++s) {
            float xt = xs[s];

            // publish h (row-major h[batch][k]) for the B-operand transpose
            #pragma unroll
            for (int p = 0; p < 4; ++p) {
                v2f hv; hv.x = h[2 * p]; hv.y = h[2 * p + 1];
                *(v2f*)(hbuf + col * 18 + (2 * p + 8 * half)) = hv;
            }
            __syncthreads();
            // B slice j, reg p, lane l: h[l%16][4j + p + 2*(l/16)]
            v2f Bm[4];
            #pragma unroll
            for (int j = 0; j < 4; ++j)
                Bm[j] = *(const v2f*)(hbuf + col * 18 + 4 * j + 2 * half);
            __syncthreads();

            // three gate chains: acc_g = W_hh_g * h + bias_g
            v8f accR = {cr0[0], cr0[1], cr0[2], cr0[3], cr0[4], cr0[5], cr0[6], cr0[7]};
            v8f accZ = {cz0[0], cz0[1], cz0[2], cz0[3], cz0[4], cz0[5], cz0[6], cz0[7]};
            v8f accN = {cn0[0], cn0[1], cn0[2], cn0[3], cn0[4], cn0[5], cn0[6], cn0[7]};
            #pragma unroll
            for (int j = 0; j < 4; ++j) {
                accR = __builtin_amdgcn_wmma_f32_16x16x4_f32(
                    false, A[0][j], false, Bm[j], (short)0, accR, false, false);
                accZ = __builtin_amdgcn_wmma_f32_16x16x4_f32(
                    false, A[1][j], false, Bm[j], (short)0, accZ, false, false);
                accN = __builtin_amdgcn_wmma_f32_16x16x4_f32(
                    false, A[2][j], false, Bm[j], (short)0, accN, false, false);
            }

            // gates + state update, all in C/D layout (no shuffles needed)
            #pragma unroll
            for (int v = 0; v < 8; ++v) {
                float r  = sigmoid_fast(accR[v] + xt * wr[v]);
                float z  = sigmoid_fast(accZ[v] + xt * wz[v]);
                float xn = xt * wn[v] + bnc[v];
                float nn = tanh_fast(xn + r * accN[v]);
                h[v] = nn + z * (h[v] - nn);
            }
        }
    }

    // ---- fc: out[b] = sum_n h[b][n] * fc_w[n] + fc_b ----
    float partial = 0.0f;
    #pragma unroll
    for (int v = 0; v < 8; ++v) partial += h[v] * fw[v];
    red[lane] = partial;
    __syncthreads();
    if (lane < 16)
        out[b0 + lane] = partial + red[lane + 16] + fc_b[0];
}

extern "C" void kernel_launch(void* const* d_in, const int* in_sizes, int n_in,
                              void* d_out, int out_size, void* d_ws, size_t ws_size,
                              hipStream_t stream) {
    const float* x    = (const float*)d_in[0];
    const float* W_ih = (const float*)d_in[1];
    const float* W_hh = (const float*)d_in[2];
    const float* b_ih = (const float*)d_in[3];
    const float* b_hh = (const float*)d_in[4];
    const float* fc_w = (const float*)d_in[5];
    const float* fc_b = (const float*)d_in[6];
    float* out = (float*)d_out;

    const int B = in_sizes[0] / GRU_T;   // 8192 (x has B*T*1 elements)
    gru_wmma_kernel<<<B / 16, 32, 0, stream>>>(x, W_ih, W_hh, b_ih, b_hh,
                                               fc_w, fc_b, out);
}